// ChebConv_62405874811912
// MI455X (gfx1250) — compile-verified
//
#include <hip/hip_runtime.h>
#include <stdint.h>

#define DFEAT 64      // feature dim (fixed by reference)
#define DV4   16      // DFEAT / 4 float4 chunks per row
#define EPB   64      // edges staged per block in scatter kernel
#define TPB   256     // threads per block (8 wave32 waves)

typedef unsigned int u32;
typedef unsigned long long u64;
typedef u32 u32x4_t __attribute__((ext_vector_type(4)));
typedef int i32x4_t __attribute__((ext_vector_type(4)));
typedef int i32x8_t __attribute__((ext_vector_type(8)));

#if __has_builtin(__builtin_amdgcn_tensor_load_to_lds) && \
    __has_builtin(__builtin_amdgcn_s_wait_tensorcnt)
#define HAVE_TDM 1
#else
#define HAVE_TDM 0
#endif

__device__ __forceinline__ float relu_f(float x) { return fmaxf(x, 0.0f); }

#if HAVE_TDM
// 1-D TDM load: copy tile int32s from gsrc into LDS at byte offset lds_off.
// tensor_dim0 = #valid elements (OOB reads return zero -> tail-safe).
// Descriptor layout per CDNA5 ISA.
__device__ __forceinline__ void tdm_load_i32_tile(const int* gsrc, u32 lds_off,
                                                  u32 valid, u32 tile) {
    u64 ga = (u64)(uintptr_t)gsrc;
    u32x4_t g0;
    g0.x = 1u;                                            // count=1, user D#
    g0.y = lds_off;                                       // LDS byte address
    g0.z = (u32)ga;                                       // global_addr[31:0]
    g0.w = (u32)((ga >> 32) & 0x01FFFFFFu) | (2u << 30);  // addr[56:32] | type=2
    i32x8_t g1;
    g1[0] = (int)(2u << 16);                  // wg_mask=0, data_size=2 (4B)
    g1[1] = (int)((valid & 0xFFFFu) << 16);   // tensor_dim0[15:0]  (bits 63:48)
    g1[2] = (int)(((valid >> 16) & 0xFFFFu)   // tensor_dim0[31:16] (bits 79:64)
                  | (1u << 16));              // tensor_dim1 = 1    (bits 95:80)
    g1[3] = (int)(tile << 16);                // tile_dim0          (bits 127:112)
    g1[4] = 1;                                // tile_dim1=1, tile_dim2=0
    g1[5] = (int)tile;                        // tensor_dim0_stride lo (unused, dim1==1)
    g1[6] = 0;                                // stride hi / dim1_stride lo
    g1[7] = 0;                                // dim1_stride hi
    i32x4_t z4 = {0, 0, 0, 0};
#if __has_include(<hip/amd_detail/amd_gfx1250_TDM.h>)
    i32x8_t z8 = {0, 0, 0, 0, 0, 0, 0, 0};    // amdgpu-toolchain: 6-arg builtin
    __builtin_amdgcn_tensor_load_to_lds(g0, g1, z4, z4, z8, 0);
#else
    __builtin_amdgcn_tensor_load_to_lds(g0, g1, z4, z4, 0);  // ROCm 7.2: 5-arg
#endif
}
#endif

// ---------------- kernel 0: zero deg[N] and H[N*D] -------------------------
__global__ void cheb_init_kernel(float* __restrict__ deg, float* __restrict__ H,
                                 int N, int ND) {
    int i = blockIdx.x * blockDim.x + threadIdx.x;
    if (i < ND) H[i] = 0.0f;
    if (i < N)  deg[i] = 0.0f;
}

// ---------------- kernel 1: degree via fp atomics --------------------------
__global__ void cheb_deg_kernel(const int* __restrict__ dst, float* __restrict__ deg,
                                int E) {
    int e = blockIdx.x * blockDim.x + threadIdx.x;
    if (e < E) {
        int d = __builtin_nontemporal_load(dst + e);
        unsafeAtomicAdd(&deg[d], 1.0f);
    }
}

// ---------------- kernel 2: dinv = rsqrt(max(deg,1)); out[:,0:64]=relu(X0) -
__global__ void cheb_dinv_relu0_kernel(const float* __restrict__ feat,
                                       const float* __restrict__ deg,
                                       float* __restrict__ dinv,
                                       float* __restrict__ out, int N) {
    int j = blockIdx.x * blockDim.x + threadIdx.x;
    if (j >= N * DFEAT) return;
    int n = j >> 6, f = j & 63;
    if (f == 0) {
        float d = deg[n];
        d = d < 1.0f ? 1.0f : d;
        dinv[n] = rsqrtf(d);
    }
    out[(size_t)n * 192 + f] = relu_f(feat[j]);
}

// ---------------- kernel 3/5: edge scatter  H[dst] += X[src]*dinv[src] -----
// Each block handles EPB=64 edges. The 64 src + 64 dst indices are staged
// into LDS by the Tensor Data Mover (TENSORcnt path), keeping the VMEM
// return path free for the random float4 gathers and f32 atomics that
// dominate this kernel. TDM ignores EXEC, so only wave 0 issues the DMAs.
__global__ void cheb_scatter_kernel(const float4* __restrict__ X4,
                                    const int* __restrict__ src,
                                    const int* __restrict__ dst,
                                    const float* __restrict__ dinv,
                                    float4* __restrict__ H4, int E) {
    __shared__ int s_idx[2 * EPB];   // [0..63]=src indices, [64..127]=dst indices
    const int t = threadIdx.x;
    const int ebase = blockIdx.x * EPB;
    const int nval = min(EPB, E - ebase);

#if HAVE_TDM
    if (t == 0) {
        // Flat shared-aperture addresses carry the workgroup-relative LDS
        // byte offset in their low 32 bits (CDNA5 ISA aperture mapping).
        u32 base = (u32)(uintptr_t)(const void*)s_idx;
        tdm_load_i32_tile(src + ebase, base,                 (u32)nval, EPB);
        tdm_load_i32_tile(dst + ebase, base + (u32)(EPB * 4), (u32)nval, EPB);
        __builtin_amdgcn_s_wait_tensorcnt(0);
    }
    // The TDM builtin writes LDS through a raw offset, which the optimizer
    // cannot see. Make this asm a potential writer of s_idx so the ds_load
    // reads below are not folded to undef ("LDS never stored" elimination).
    asm volatile("" :: "v"(s_idx) : "memory");
#else
    if (t < EPB) {
        if (t < nval) s_idx[t] = __builtin_nontemporal_load(src + ebase + t);
    } else if (t < 2 * EPB) {
        int k = t - EPB;
        if (k < nval) s_idx[EPB + k] = __builtin_nontemporal_load(dst + ebase + k);
    }
#endif
    __syncthreads();

    // 64 edges x 16 float4 chunks = 1024 work items over 256 threads (4 iters).
    // Consecutive threads cover consecutive chunks of one row -> 256B coalesced.
    for (int it = 0; it < (EPB * DV4) / TPB; ++it) {
        int p  = it * TPB + t;
        int el = p >> 4;        // edge within tile
        int q  = p & 15;        // float4 chunk within row
        if (el >= nval) continue;
        int s  = s_idx[el];
        int d  = s_idx[EPB + el];
        float  sc = dinv[s];                          // L1/L2-resident scalar
        float4 v  = X4[(size_t)s * DV4 + q];          // 256B/row coalesced gather
        float* hp = (float*)(H4 + (size_t)d * DV4 + q);
        unsafeAtomicAdd(hp + 0, v.x * sc);            // global_atomic_add_f32 (L2)
        unsafeAtomicAdd(hp + 1, v.y * sc);
        unsafeAtomicAdd(hp + 2, v.z * sc);
        unsafeAtomicAdd(hp + 3, v.w * sc);
    }
}

// ---------------- kernel 4: X1 = (rn-1)*X0 - rn*(dinv*H); re-zero H --------
__global__ void cheb_combine1_kernel(const float* __restrict__ feat,
                                     const float* __restrict__ dinv,
                                     const float* __restrict__ lambda_max,
                                     float* __restrict__ H,
                                     float* __restrict__ X1,
                                     float* __restrict__ out, int N) {
    int j = blockIdx.x * blockDim.x + threadIdx.x;
    if (j >= N * DFEAT) return;
    int n = j >> 6, f = j & 63;
    float rn = 2.0f / lambda_max[0];
    float h  = dinv[n] * H[j];
    float x1 = (rn - 1.0f) * feat[j] - rn * h;
    X1[j] = x1;
    out[(size_t)n * 192 + 64 + f] = relu_f(x1);
    H[j] = 0.0f;   // ready for second scatter pass
}

// ---------------- kernel 6: X2 = 2(rn-1)*X1 - 2rn*(dinv*H) - X0 ------------
__global__ void cheb_combine2_kernel(const float* __restrict__ feat,
                                     const float* __restrict__ X1,
                                     const float* __restrict__ dinv,
                                     const float* __restrict__ lambda_max,
                                     const float* __restrict__ H,
                                     float* __restrict__ out, int N) {
    int j = blockIdx.x * blockDim.x + threadIdx.x;
    if (j >= N * DFEAT) return;
    int n = j >> 6, f = j & 63;
    float rn = 2.0f / lambda_max[0];
    float h  = dinv[n] * H[j];
    float x2 = 2.0f * (rn - 1.0f) * X1[j] - 2.0f * rn * h - feat[j];
    out[(size_t)n * 192 + 128 + f] = relu_f(x2);
}

extern "C" void kernel_launch(void* const* d_in, const int* in_sizes, int n_in,
                              void* d_out, int out_size, void* d_ws, size_t ws_size,
                              hipStream_t stream) {
    const float* feat = (const float*)d_in[0];
    const int*   src  = (const int*)d_in[1];
    const int*   dst  = (const int*)d_in[2];
    const float* lam  = (const float*)d_in[3];
    float*       out  = (float*)d_out;

    const int N  = in_sizes[0] / DFEAT;   // 60000
    const int E  = in_sizes[1];           // 1200000
    const int ND = N * DFEAT;

    // Workspace layout (floats): deg[N] | dinv[N] | H[N*D] | X1[N*D]  (~31 MB)
    float* w    = (float*)d_ws;
    float* deg  = w;
    float* dinv = w + N;
    size_t hoff = ((size_t)(2 * N) + 3) & ~(size_t)3;   // 16B-align H
    float* H    = w + hoff;
    float* X1   = H + (size_t)ND;

    dim3 blk(TPB);
    dim3 gND((ND + TPB - 1) / TPB);
    dim3 gE ((E + TPB - 1) / TPB);
    dim3 gSc((E + EPB - 1) / EPB);

    cheb_init_kernel      <<<gND, blk, 0, stream>>>(deg, H, N, ND);
    cheb_deg_kernel       <<<gE,  blk, 0, stream>>>(dst, deg, E);
    cheb_dinv_relu0_kernel<<<gND, blk, 0, stream>>>(feat, deg, dinv, out, N);
    cheb_scatter_kernel   <<<gSc, blk, 0, stream>>>((const float4*)feat, src, dst,
                                                    dinv, (float4*)H, E);
    cheb_combine1_kernel  <<<gND, blk, 0, stream>>>(feat, dinv, lam, H, X1, out, N);
    cheb_scatter_kernel   <<<gSc, blk, 0, stream>>>((const float4*)X1, src, dst,
                                                    dinv, (float4*)H, E);
    cheb_combine2_kernel  <<<gND, blk, 0, stream>>>(feat, X1, dinv, lam, H, out, N);
}